// QuantumLayer_9311489098334
// MI455X (gfx1250) — compile-verified
//
#include <hip/hip_runtime.h>
#include <math.h>

// Quantum layer: 10 qubits, batch 16384, depth 2.
// Per-sample 1024-amp complex state kept entirely on-chip (LDS + VGPRs).
// Layer-1 Rot application = two 32x32 complex GEMMs per sample, done with
// V_WMMA_F32_16X16X4_F32 (f32 matrix pipe, matches complex64 reference).

#define NQ 10
#define WAVES_PER_BLOCK 4

typedef float v2f __attribute__((ext_vector_type(2)));
typedef float v8f __attribute__((ext_vector_type(8)));

__device__ __forceinline__ v8f wmma4(v2f a, v2f b, v8f c) {
  // D(16x16,f32) = A(16x4,f32) x B(4x16,f32) + C
  return __builtin_amdgcn_wmma_f32_16x16x4_f32(
      false, a, false, b, (short)0, c, false, false);
}

// ---------------------------------------------------------------------------
// Precompute: A_l = Rot(l,0)⊗..⊗Rot(l,4), B_l = Rot(l,5)⊗..⊗Rot(l,9)
// (32x32 complex each, shared across batch) + CNOT-ring permutation tables.
// ---------------------------------------------------------------------------
__global__ void precompute_kernel(const float* __restrict__ qw,
                                  float2* __restrict__ mats,  // 4 * 1024 complex
                                  int* __restrict__ perms)    // 2 * 1024
{
  int tid = threadIdx.x;
  // --- Kronecker-product layer matrices ---
  for (int e = tid; e < 4096; e += blockDim.x) {
    int mat = e >> 10;          // 0:A0 1:B0 2:A1 3:B1
    int idx = e & 1023;
    int r = idx >> 5, k = idx & 31;
    int l = mat >> 1;
    int wbase = (mat & 1) * 5;
    float vr = 1.f, vi = 0.f;
    for (int j = 0; j < 5; ++j) {
      int q = wbase + j;
      const float* w = qw + (l * NQ + q) * 3;
      float phi = w[0], theta = w[1], omega = w[2];
      float ch = cosf(theta * 0.5f), sh = sinf(theta * 0.5f);
      int bi = (r >> (4 - j)) & 1;   // wire (wbase+j) is MSB-first within group
      int bj = (k >> (4 - j)) & 1;
      float ang, mag;
      if (!bi && !bj)      { ang = -0.5f * (phi + omega); mag =  ch; }
      else if (!bi &&  bj) { ang =  0.5f * (phi - omega); mag = -sh; }
      else if ( bi && !bj) { ang = -0.5f * (phi - omega); mag =  sh; }
      else                 { ang =  0.5f * (phi + omega); mag =  ch; }
      float er = mag * cosf(ang), ei = mag * sinf(ang);
      float nr = vr * er - vi * ei;
      float ni = vr * ei + vi * er;
      vr = nr; vi = ni;
    }
    mats[mat * 1024 + idx] = make_float2(vr, vi);
  }
  // --- CNOT-ring permutations (range r = l+1), sequential composition ---
  for (int i = tid; i < 1024; i += blockDim.x) {
    for (int l = 0; l < 2; ++l) {
      int rr = l + 1;
      int b[NQ];
      for (int q = 0; q < NQ; ++q) b[q] = (i >> (NQ - 1 - q)) & 1;
      for (int q = 0; q < NQ; ++q) b[(q + rr) % NQ] ^= b[q];
      int dest = 0;
      for (int q = 0; q < NQ; ++q) dest |= b[q] << (NQ - 1 - q);
      perms[l * 1024 + i] = dest;
    }
  }
}

// ---------------------------------------------------------------------------
// Main kernel: one wave32 = one sample. State = 32x32 complex tile in LDS.
// ---------------------------------------------------------------------------
__global__ __launch_bounds__(32 * WAVES_PER_BLOCK)
void qlayer_kernel(const float* __restrict__ x,
                   const float2* __restrict__ mats,
                   const int* __restrict__ perms,
                   float* __restrict__ out)
{
  __shared__ float2 lds_all[WAVES_PER_BLOCK * 1024];  // 32 KB
  const int lane = threadIdx.x & 31;
  const int wave = threadIdx.x >> 5;
  float2* S = lds_all + wave * 1024;
  const int b = blockIdx.x * WAVES_PER_BLOCK + wave;

  const float2* A0 = mats;
  const float2* B0 = mats + 1024;
  const float2* A1 = mats + 2048;
  const float2* B1 = mats + 3072;
  const int* perm1 = perms;
  const int* perm2 = perms + 1024;

  // ---- embedding: product state => rank-1 outer product u v^T (real) ----
  float cq[NQ], sq[NQ];
  #pragma unroll
  for (int q = 0; q < NQ; ++q) {
    float xv = x[b * NQ + q];
    cq[q] = cosf(0.5f * xv);
    sq[q] = sinf(0.5f * xv);
  }
  float u = 1.f, v = 1.f;
  #pragma unroll
  for (int j = 0; j < 5; ++j) {
    int bit = (lane >> (4 - j)) & 1;
    u *= bit ? sq[j] : cq[j];
    v *= bit ? sq[5 + j] : cq[5 + j];
  }

  // ---- layer 0 on rank-1 state: a = A0 u, bb = B0 v (complex matvecs) ----
  float ar_ = 0.f, ai_ = 0.f, br_ = 0.f, bi_ = 0.f;
  for (int k = 0; k < 32; ++k) {
    float uk = __shfl(u, k, 32);
    float vk = __shfl(v, k, 32);
    float2 Ae = A0[lane * 32 + k];
    float2 Be = B0[lane * 32 + k];
    ar_ += Ae.x * uk; ai_ += Ae.y * uk;
    br_ += Be.x * vk; bi_ += Be.y * vk;
  }
  // ---- CNOT ring r=1: scatter outer product a * bb^T through perm1 ----
  for (int r = 0; r < 32; ++r) {
    float xr = __shfl(ar_, r, 32);
    float xi = __shfl(ai_, r, 32);
    float wr = xr * br_ - xi * bi_;
    float wi = xr * bi_ + xi * br_;
    int d = perm1[r * 32 + lane];
    S[d] = make_float2(wr, wi);
  }
  __syncthreads();

  const v8f vzero = {0.f, 0.f, 0.f, 0.f, 0.f, 0.f, 0.f, 0.f};
  const int mh = lane & 15;            // A-op row / B-op col within a tile
  const int kh = lane >> 4;            // K-half selector
  const int mbase = kh * 8;            // C/D tile: lanes 16-31 hold M=8..15
  const int nn = lane & 15;            // C/D tile column

  // ---- stage 1: P = A1 @ S  (complex, WMMA f32 16x16x4) ----
  v8f accR[2][2], accI[2][2];
  #pragma unroll
  for (int I = 0; I < 2; ++I)
    #pragma unroll
    for (int J = 0; J < 2; ++J) { accR[I][J] = vzero; accI[I][J] = vzero; }

  #pragma unroll
  for (int I = 0; I < 2; ++I) {
    #pragma unroll
    for (int J = 0; J < 2; ++J) {
      #pragma unroll
      for (int kk = 0; kk < 32; kk += 4) {
        int k0 = kk + 2 * kh;
        float2 a0 = A1[(16 * I + mh) * 32 + k0];
        float2 a1 = A1[(16 * I + mh) * 32 + k0 + 1];
        v2f aR = {a0.x, a1.x};
        v2f aI = {a0.y, a1.y};
        v2f aIn = {-a0.y, -a1.y};
        float2 s0 = S[k0 * 32 + 16 * J + mh];
        float2 s1 = S[(k0 + 1) * 32 + 16 * J + mh];
        v2f bR = {s0.x, s1.x};
        v2f bI = {s0.y, s1.y};
        accR[I][J] = wmma4(aR,  bR, accR[I][J]);
        accR[I][J] = wmma4(aIn, bI, accR[I][J]);
        accI[I][J] = wmma4(aR,  bI, accI[I][J]);
        accI[I][J] = wmma4(aI,  bR, accI[I][J]);
      }
    }
  }
  __syncthreads();
  // write P back to LDS (C/D layout -> natural row-major complex tile)
  #pragma unroll
  for (int I = 0; I < 2; ++I)
    #pragma unroll
    for (int J = 0; J < 2; ++J)
      #pragma unroll
      for (int t = 0; t < 8; ++t) {
        int rr = 16 * I + mbase + t;
        int cc = 16 * J + nn;
        S[rr * 32 + cc] = make_float2(accR[I][J][t], accI[I][J][t]);
      }
  __syncthreads();

  // ---- stage 2: Q = P @ B1^T (complex, WMMA) ----
  v8f qR[2][2], qI[2][2];
  #pragma unroll
  for (int I = 0; I < 2; ++I)
    #pragma unroll
    for (int J = 0; J < 2; ++J) { qR[I][J] = vzero; qI[I][J] = vzero; }

  #pragma unroll
  for (int I = 0; I < 2; ++I) {
    #pragma unroll
    for (int J = 0; J < 2; ++J) {
      #pragma unroll
      for (int kk = 0; kk < 32; kk += 4) {
        int k0 = kk + 2 * kh;
        float2 p0 = S[(16 * I + mh) * 32 + k0];
        float2 p1 = S[(16 * I + mh) * 32 + k0 + 1];
        v2f pR = {p0.x, p1.x};
        v2f pI = {p0.y, p1.y};
        v2f pIn = {-p0.y, -p1.y};
        // B-operand element (k, n) of B1^T = B1[16J+n][k]
        float2 b0 = B1[(16 * J + mh) * 32 + k0];
        float2 b1 = B1[(16 * J + mh) * 32 + k0 + 1];
        v2f bR = {b0.x, b1.x};
        v2f bI = {b0.y, b1.y};
        qR[I][J] = wmma4(pR,  bR, qR[I][J]);
        qR[I][J] = wmma4(pIn, bI, qR[I][J]);
        qI[I][J] = wmma4(pR,  bI, qI[I][J]);
        qI[I][J] = wmma4(pI,  bR, qI[I][J]);
      }
    }
  }
  __syncthreads();
  // ---- CNOT ring r=2: scatter Q through perm2 ----
  #pragma unroll
  for (int I = 0; I < 2; ++I)
    #pragma unroll
    for (int J = 0; J < 2; ++J)
      #pragma unroll
      for (int t = 0; t < 8; ++t) {
        int idx = (16 * I + mbase + t) * 32 + (16 * J + nn);
        int d = perm2[idx];
        S[d] = make_float2(qR[I][J][t], qI[I][J][t]);
      }
  __syncthreads();

  // ---- PauliZ expvals: E[q] = sum_i |amp_i|^2 * (1 - 2*bit_q(i)) ----
  float e[NQ];
  #pragma unroll
  for (int q = 0; q < NQ; ++q) e[q] = 0.f;
  for (int row = 0; row < 32; ++row) {
    int i = row * 32 + lane;
    float2 amp = S[i];
    float p = amp.x * amp.x + amp.y * amp.y;
    #pragma unroll
    for (int q = 0; q < NQ; ++q)
      e[q] += ((i >> (NQ - 1 - q)) & 1) ? -p : p;
  }
  #pragma unroll
  for (int q = 0; q < NQ; ++q) {
    #pragma unroll
    for (int off = 16; off > 0; off >>= 1)
      e[q] += __shfl_xor(e[q], off, 32);
  }
  if (lane == 0) {
    #pragma unroll
    for (int q = 0; q < NQ; ++q) out[b * NQ + q] = e[q];
  }
}

// ---------------------------------------------------------------------------
extern "C" void kernel_launch(void* const* d_in, const int* in_sizes, int n_in,
                              void* d_out, int out_size, void* d_ws, size_t ws_size,
                              hipStream_t stream) {
  const float* x  = (const float*)d_in[0];   // [B, 10] f32
  const float* qw = (const float*)d_in[1];   // [2, 10, 3] f32
  float* out = (float*)d_out;                // [B, 10] f32

  float2* mats = (float2*)d_ws;                               // 4*1024 complex (32 KB)
  int* perms = (int*)((char*)d_ws + 4 * 1024 * sizeof(float2)); // 2*1024 ints (8 KB)

  hipLaunchKernelGGL(precompute_kernel, dim3(1), dim3(256), 0, stream,
                     qw, mats, perms);

  int batch = in_sizes[0] / NQ;
  hipLaunchKernelGGL(qlayer_kernel, dim3(batch / WAVES_PER_BLOCK),
                     dim3(32 * WAVES_PER_BLOCK), 0, stream,
                     x, mats, perms, out);
}